// SimilarUserRetriever_63221918597187
// MI455X (gfx1250) — compile-verified
//
#include <hip/hip_runtime.h>
#include <math.h>

// ---------------------------------------------------------------------------
// SimilarUserRetriever for MI455X (gfx1250, wave32, WMMA + async-LDS DMA)
//   out = (similar_embeddings [B,5,128], similarity_scores [B,5]) flat-concat
// ---------------------------------------------------------------------------

constexpr int D      = 128;
constexpr int TOPK   = 5;
constexpr int K1     = 6;          // TOPK + 1 candidates before dedup
constexpr int QB     = 32;         // queries per block (2 WMMA M-tiles)
constexpr int SUPER  = 64;         // users per LDS supertile (4 WMMA N-tiles)
constexpr int LDST   = 132;        // padded LDS row stride (floats): banks spread
constexpr float DUP  = 0.9999f;

typedef float v2f __attribute__((ext_vector_type(2)));
typedef float v8f __attribute__((ext_vector_type(8)));

// ---------------------------------------------------------------------------
// Kernel 1: inverse L2 norms for user table and query batch.
// One wave32 per row of 128 floats (float4 per lane + xor-shuffle reduce).
// ---------------------------------------------------------------------------
__global__ __launch_bounds__(256)
void norms_kernel(const float* __restrict__ users,
                  const float* __restrict__ queries,
                  float* __restrict__ uinv, float* __restrict__ qinv,
                  int n_users, int n_queries)
{
    int wave = threadIdx.x >> 5;
    int lane = threadIdx.x & 31;
    int row  = blockIdx.x * 8 + wave;
    int total = n_users + n_queries;
    if (row >= total) return;

    const float* src = (row < n_users) ? (users  + (size_t)row * D)
                                       : (queries + (size_t)(row - n_users) * D);
    float4 v = ((const float4*)src)[lane];
    float s = v.x * v.x + v.y * v.y + v.z * v.z + v.w * v.w;
#pragma unroll
    for (int m = 16; m > 0; m >>= 1) s += __shfl_xor(s, m, 32);

    if (lane == 0) {
        float inv = 1.0f / fmaxf(sqrtf(s), 1e-12f);
        if (row < n_users) uinv[row] = inv;
        else               qinv[row - n_users] = inv;
    }
}

// ---------------------------------------------------------------------------
// Kernel 2: fused (q_norm @ ue_norm^T) + streaming top-6 + dedup + gather.
// Double-buffered user tiles DMA'd straight into LDS with
// global_load_async_to_lds_b128 (ASYNCcnt), overlapping copy with WMMA.
// User-norm scaling is applied in the C-tile epilogue (column-constant).
// ---------------------------------------------------------------------------
__global__ __launch_bounds__(256)
void simtopk_kernel(const float* __restrict__ users,
                    const float* __restrict__ queries,
                    const float* __restrict__ uinv,
                    const float* __restrict__ qinv,
                    float* __restrict__ out_emb,   // [B,5,128]
                    float* __restrict__ out_sc,    // [B,5]
                    int n_users)
{
    __shared__ float Us[2][SUPER * LDST];  // 67584 B, double-buffered raw users
    __shared__ float qs[QB * LDST];        // 16896 B, queries pre-scaled by qinv
    __shared__ float scr[8][16][17];       //  8704 B, per-wave C-tile transpose

    const int t     = threadIdx.x;
    const int wave  = t >> 5;
    const int lane  = t & 31;
    const int m     = lane & 15;           // M (A) / N (B,C) within tile
    const int hi    = lane >> 4;           // K-half selector per ISA layout
    const int qtile = wave >> 2;           // 0..1
    const int sub   = wave & 3;            // user subtile 0..3
    const int qb    = blockIdx.x * QB;

    // ---- issue one supertile of async global->LDS copies (8 x b128/thread) --
    auto issue_tile = [&](int sblk, int buf) {
        int base = sblk * SUPER;
#pragma unroll
        for (int i = 0; i < 8; ++i) {
            int j   = t + i * 256;
            int row = j >> 5;              // 32 float4 per user row
            int c4  = j & 31;
            int u   = base + row;
            if (u > n_users - 1) u = n_users - 1;   // clamp: no OOB DMA reads
            const float* g = users + (size_t)u * D + c4 * 4;
            unsigned lds = (unsigned)(size_t)(&Us[buf][row * LDST + c4 * 4]);
            asm volatile("global_load_async_to_lds_b128 %0, %1, off"
                         :: "v"(lds), "v"((unsigned long long)(size_t)g)
                         : "memory");
        }
    };

    // ---- stage normalized queries into LDS (QB*D = 1024 float4, 4/thread) ----
#pragma unroll
    for (int i = 0; i < 4; ++i) {
        int j   = t + i * 256;
        int row = j >> 5;
        int c4  = j & 31;
        float4 v = ((const float4*)(queries + (size_t)(qb + row) * D))[c4];
        float  s = qinv[qb + row];
        float4 w = make_float4(v.x * s, v.y * s, v.z * s, v.w * s);
        *(float4*)(qs + row * LDST + c4 * 4) = w;   // 132%4==0 -> 16B aligned
    }

    // ---- per-lane running top-6 (lanes 0..15 own one query row each) ----
    float tv[K1]; int ti[K1];
#pragma unroll
    for (int i = 0; i < K1; ++i) { tv[i] = -3.0e38f; ti[i] = 0; }

    const float* arow = qs + (qtile * 16 + m) * LDST;
    const int nsuper = (n_users + SUPER - 1) / SUPER;

    issue_tile(0, 0);                      // prime the pipeline

    for (int sblk = 0; sblk < nsuper; ++sblk) {
        const int base = sblk * SUPER;
        const int buf  = sblk & 1;

        // my DMA for tile sblk is done; barrier also guarantees every wave
        // finished computing on the other buffer (end of previous iteration).
        asm volatile("s_wait_asynccnt 0x0" ::: "memory");
        __syncthreads();

        if (sblk + 1 < nsuper) issue_tile(sblk + 1, buf ^ 1);

        // ---- 16x16 similarity tile via f32 WMMA, K = 128 in 32 steps ----
        const float* brow = &Us[buf][(sub * 16 + m) * LDST];
        v8f c = {};
#pragma unroll
        for (int kc = 0; kc < 32; ++kc) {
            int k0 = kc * 4 + 2 * hi;                 // ISA A/B K mapping
            v2f a = *(const v2f*)(arow + k0);         // 8B aligned (k0 even)
            v2f b = *(const v2f*)(brow + k0);
            c = __builtin_amdgcn_wmma_f32_16x16x4_f32(
                    false, a, false, b, (short)0, c, false, false);
        }

        // ---- epilogue: scale by user inverse norm (column-constant), then
        //      transpose C through LDS: scr[wave][M][N] ----
        int un   = base + sub * 16 + m;
        float su = (un < n_users) ? uinv[un] : 0.0f;
#pragma unroll
        for (int r = 0; r < 8; ++r)
            scr[wave][r + 8 * hi][m] = c[r] * su;
        __syncthreads();

        // ---- streaming top-6 insert (lane q handles its query row) ----
        if (lane < 16) {
            for (int j = 0; j < 16; ++j) {
                int u = base + sub * 16 + j;
                if (u >= n_users) break;              // ragged tail mask
                float s = scr[wave][lane][j];
                if (s > tv[K1 - 1]) {
                    tv[K1 - 1] = s; ti[K1 - 1] = u;
#pragma unroll
                    for (int p = K1 - 1; p > 0; --p) {
                        if (tv[p] > tv[p - 1]) {
                            float tf = tv[p]; tv[p] = tv[p - 1]; tv[p - 1] = tf;
                            int   tn = ti[p]; ti[p] = ti[p - 1]; ti[p - 1] = tn;
                        }
                    }
                }
            }
        }
    }

    __syncthreads();   // done with Us -> reuse as merge arena

    float* arena = &Us[0][0];
    float (*msc)[4 * K1] = (float (*)[4 * K1])arena;                 // [QB][24]
    int   (*mix)[4 * K1] = (int   (*)[4 * K1])(arena + QB * 4 * K1); // [QB][24]
    int   (*sel)[TOPK]   = (int   (*)[TOPK])  (arena + 2 * QB * 4 * K1);

    if (lane < 16) {
#pragma unroll
        for (int e = 0; e < K1; ++e) {
            msc[qtile * 16 + lane][sub * K1 + e] = tv[e];
            mix[qtile * 16 + lane][sub * K1 + e] = ti[e];
        }
    }
    __syncthreads();

    // ---- merge 24 candidates -> top-6 -> dedup/pad (reference-exact) ----
    if (t < QB) {
        float bv[K1]; int bi[K1];
        unsigned used = 0;
#pragma unroll
        for (int s = 0; s < K1; ++s) {
            float best = -3.4e38f; int bj = 0;
            for (int j = 0; j < 4 * K1; ++j) {
                bool ok = !((used >> j) & 1u) && (msc[t][j] > best);
                if (ok) { best = msc[t][j]; bj = j; }
            }
            used |= 1u << bj;
            bv[s] = best; bi[s] = mix[t][bj];
        }
        // mask >= 0.9999, take first 5 valid, pad with last valid, fallback raw
        float vv[TOPK]; int vi[TOPK]; int nv = 0;
#pragma unroll
        for (int s = 0; s < K1; ++s) {
            if (bv[s] < DUP) {
                if (nv < TOPK) { vv[nv] = bv[s]; vi[nv] = bi[s]; }
                nv++;
            }
        }
        int q_global = qb + t;
        int lim = (nv < TOPK ? nv : TOPK) - 1;
#pragma unroll
        for (int s = 0; s < TOPK; ++s) {
            float val; int idx;
            if (nv > 0) { int p = s < lim ? s : lim; val = vv[p]; idx = vi[p]; }
            else        { val = bv[s]; idx = bi[s]; }
            out_sc[(size_t)q_global * TOPK + s] = val;
            sel[t][s] = idx;
        }
    }
    __syncthreads();

    // ---- gather normalized embeddings: 5120 float4, 20/thread ----
#pragma unroll
    for (int i = 0; i < 20; ++i) {
        int j    = t + i * 256;
        int slot = j >> 5;                   // (q, s) pair, 32 float4 per row
        int q    = slot / TOPK;
        int s    = slot % TOPK;
        int c4   = j & 31;
        int u    = sel[q][s];
        float sc = uinv[u];
        float4 v = ((const float4*)(users + (size_t)u * D))[c4];
        float4 w = make_float4(v.x * sc, v.y * sc, v.z * sc, v.w * sc);
        *(float4*)(out_emb + ((size_t)(qb + q) * TOPK + s) * D + c4 * 4) = w;
    }
}

// ---------------------------------------------------------------------------
extern "C" void kernel_launch(void* const* d_in, const int* in_sizes, int n_in,
                              void* d_out, int out_size, void* d_ws, size_t ws_size,
                              hipStream_t stream)
{
    const float* q = (const float*)d_in[0];   // [B,128]
    const float* u = (const float*)d_in[1];   // [N,128]
    const int B = in_sizes[0] / D;            // 4096
    const int N = in_sizes[1] / D;            // 100000

    float* wsf  = (float*)d_ws;
    float* uinv = wsf;                        // [N]
    float* qinv = wsf + N;                    // [B]

    float* out_emb = (float*)d_out;                      // [B,5,128]
    float* out_sc  = out_emb + (size_t)B * TOPK * D;     // [B,5]

    int rows = N + B;
    norms_kernel<<<(rows + 7) / 8, 256, 0, stream>>>(u, q, uinv, qinv, N, B);
    simtopk_kernel<<<B / QB, 256, 0, stream>>>(u, q, uinv, qinv,
                                               out_emb, out_sc, N);
}